// unsliding_windows_38903813767371
// MI455X (gfx1250) — compile-verified
//
#include <hip/hip_runtime.h>
#include <stdint.h>

// Overlap-add fold, WIDTH = 2*STEP  ==>  race-free gather:
//   out[c, t] = x[t/128, c, t%128] + x[t/128 - 1, c, t%128 + 128]
// One wave32 per (channel, group-of-4 output blocks). Per wave:
//   4 KB HBM -> LDS via 8x global_load_async_to_lds_b128 (ASYNCcnt),
//   8x ds_load_b128, 16x v_add_f32, 4x global_store_b128 (2 KB out).
// Every input element is read exactly once; every output written exactly once.

namespace {
constexpr int kC = 64;     // channels
constexpr int kW = 256;    // window width (floats)
constexpr int kS = 128;    // step (floats)
constexpr int kR = 4;      // output blocks of 128 per wave
constexpr int kWaves = 8;  // waves per 256-thread block
// Per-wave LDS floats: P(128) + W0..W2(256 each) + W3 first half(128) = 1024
constexpr int kLdsPerWave = 1024;
}

__global__ __launch_bounds__(256) void fold_ola_async4_kernel(
    const float* __restrict__ x, float* __restrict__ out,
    int nwin, int nblk, long long total_len, int num_tasks)
{
    const int lane = (int)(threadIdx.x & 31u);
    const int wave = (int)(threadIdx.x >> 5u);
    const int task = (int)blockIdx.x * kWaves + wave;
    if (task >= num_tasks) return;

    const int c = task & (kC - 1);
    const int b = (task >> 6) * kR;            // first output block of this wave

    __shared__ __align__(16) float smem[kWaves * kLdsPerWave];  // 32 KB
    float* wb = &smem[wave * kLdsPerWave];

    // Workgroup-relative LDS byte address (low 32 bits of generic LDS pointer).
    const uint32_t lane16  = (uint32_t)lane * 16u;
    const uint32_t ldsBase = (uint32_t)(uintptr_t)wb + lane16;
    const uint32_t ldsP  = ldsBase;            // bytes [   0,  512): win b-1, w in [128,256)
    const uint32_t ldsW0 = ldsBase + 512u;     // bytes [ 512, 1536): win b
    const uint32_t ldsW1 = ldsBase + 1536u;    // bytes [1536, 2560): win b+1
    const uint32_t ldsW2 = ldsBase + 2560u;    // bytes [2560, 3584): win b+2
    const uint32_t ldsW3 = ldsBase + 3584u;    // bytes [3584, 4096): win b+3, w in [0,128)

    // Global byte offsets into x (x spans 512 MB -> fits u32). Clamp window
    // indices so every async load is unconditional; edge contributions are
    // discarded by selects below.
    const uint32_t wstride = (uint32_t)(kC * kW * 4);  // 65536 B per window
    const uint32_t cOff    = (uint32_t)c * (kW * 4);   // 1024 B per channel
    const int wmax = nwin - 1;
    const int wP = (b > 0) ? (b - 1) : 0;
    const int w0 = (b     <= wmax) ? b     : wmax;
    const int w1 = (b + 1 <= wmax) ? b + 1 : wmax;
    const int w2 = (b + 2 <= wmax) ? b + 2 : wmax;
    const int w3 = (b + 3 <= wmax) ? b + 3 : wmax;
    const uint32_t vP = (uint32_t)wP * wstride + cOff + 512u + lane16;  // second half
    const uint32_t v0 = (uint32_t)w0 * wstride + cOff + lane16;
    const uint32_t v1 = (uint32_t)w1 * wstride + cOff + lane16;
    const uint32_t v2 = (uint32_t)w2 * wstride + cOff + lane16;
    const uint32_t v3 = (uint32_t)w3 * wstride + cOff + lane16;

    // NOTE: the instruction offset is added to BOTH the LDS and global address,
    // so each window's two 512 B halves share one (lds, voff) register pair.
    asm volatile("global_load_async_to_lds_b128 %0, %1, %2"            :: "v"(ldsP),  "v"(vP), "s"(x) : "memory");
    asm volatile("global_load_async_to_lds_b128 %0, %1, %2"            :: "v"(ldsW0), "v"(v0), "s"(x) : "memory");
    asm volatile("global_load_async_to_lds_b128 %0, %1, %2 offset:512" :: "v"(ldsW0), "v"(v0), "s"(x) : "memory");
    asm volatile("global_load_async_to_lds_b128 %0, %1, %2"            :: "v"(ldsW1), "v"(v1), "s"(x) : "memory");
    asm volatile("global_load_async_to_lds_b128 %0, %1, %2 offset:512" :: "v"(ldsW1), "v"(v1), "s"(x) : "memory");
    asm volatile("global_load_async_to_lds_b128 %0, %1, %2"            :: "v"(ldsW2), "v"(v2), "s"(x) : "memory");
    asm volatile("global_load_async_to_lds_b128 %0, %1, %2 offset:512" :: "v"(ldsW2), "v"(v2), "s"(x) : "memory");
    asm volatile("global_load_async_to_lds_b128 %0, %1, %2"            :: "v"(ldsW3), "v"(v3), "s"(x) : "memory");

    // Wait for this wave's async DMA into LDS to complete.
    asm volatile("s_wait_asynccnt 0" ::: "memory");

    // Operand streams in LDS (float4 indices off one base register):
    //   block r: current  a_r at f4 idx 32 + 64*r + lane
    //            previous p_r at f4 idx      64*r + lane  (r=0 hits P region)
    const float4* lp = (const float4*)wb;
    float4* dst = (float4*)(out + (size_t)c * (size_t)total_len + (size_t)b * kS) + lane;

    const bool interior = (b >= 1) && (b + (kR - 1) < nwin);
    if (interior) {
#pragma unroll
        for (int r = 0; r < kR; ++r) {
            const float4 a = lp[32 + 64 * r + lane];
            const float4 p = lp[64 * r + lane];
            float4 v;
            v.x = a.x + p.x; v.y = a.y + p.y; v.z = a.z + p.z; v.w = a.w + p.w;
            dst[32 * r] = v;
        }
    } else {
        // First group (b==0: no previous window for r=0) and last group
        // (b+r may exceed the last window / last output block).
#pragma unroll
        for (int r = 0; r < kR; ++r) {
            if (b + r < nblk) {
                const float4 a = lp[32 + 64 * r + lane];
                const float4 p = lp[64 * r + lane];
                const bool hA = (b + r) < nwin;  // window b+r exists
                const bool hB = (b + r) > 0;     // window b+r-1 exists
                float4 v;
                v.x = (hA ? a.x : 0.0f) + (hB ? p.x : 0.0f);
                v.y = (hA ? a.y : 0.0f) + (hB ? p.y : 0.0f);
                v.z = (hA ? a.z : 0.0f) + (hB ? p.z : 0.0f);
                v.w = (hA ? a.w : 0.0f) + (hB ? p.w : 0.0f);
                dst[32 * r] = v;
            }
        }
    }
}

extern "C" void kernel_launch(void* const* d_in, const int* in_sizes, int n_in,
                              void* d_out, int out_size, void* d_ws, size_t ws_size,
                              hipStream_t stream) {
    (void)n_in; (void)out_size; (void)d_ws; (void)ws_size;

    const float* x = (const float*)d_in[0];
    float* out = (float*)d_out;

    const long long n_elems = (long long)in_sizes[0];        // N * C * W
    const int nwin = (int)(n_elems / (long long)(kC * kW));  // N = 8192
    const long long total_len = (long long)(nwin - 1) * kS + kW;  // 1,048,704
    const int nblk = (int)(total_len / kS);                       // 8,193
    const int ngroups = (nblk + kR - 1) / kR;                     // 2,049
    const int num_tasks = ngroups * kC;                           // 131,136
    const int blocks = (num_tasks + kWaves - 1) / kWaves;         // 16,392

    fold_ola_async4_kernel<<<blocks, kWaves * 32, 0, stream>>>(
        x, out, nwin, nblk, total_len, num_tasks);
}